// DTW_Loss_79869211836822
// MI455X (gfx1250) — compile-verified
//
#include <hip/hip_runtime.h>
#include <stdint.h>

// DTW loss on MI455X (gfx1250, wave32).
// One wave32 per batch element. Lane t owns 16 columns [16t, 16t+16).
// Skewed wavefront: at step s, lane t processes DP row r = s - t.
// Cross-lane deps via __shfl_up (ds_bpermute); x staged to LDS via the
// CDNA5 async data-mover path (global_load_async_to_lds_b128 + s_wait_asynccnt).

#define DTW_B 64
#define DTW_N 512
#define COLS_PER_LANE 16   // 32 lanes * 16 = 512 columns

__device__ __forceinline__ void async_b128_to_lds(uint32_t lds_byte_addr,
                                                  const void* gaddr) {
    // GV mode: vdst = LDS byte address (per lane), vaddr = 64-bit global addr.
    asm volatile("global_load_async_to_lds_b128 %0, %1, off"
                 :: "v"(lds_byte_addr), "v"(gaddr)
                 : "memory");
}

__device__ __forceinline__ void wait_asynccnt0() {
    asm volatile("s_wait_asynccnt 0" ::: "memory");
}

__global__ void __launch_bounds__(32)
dtw_batch_kernel(const float* __restrict__ pred,    // (B, N, 2)
                 const float* __restrict__ target,  // (B, N, 2)
                 float* __restrict__ ws)            // (B,) per-batch DTW cost
{
    const int b    = blockIdx.x;
    const int lane = threadIdx.x;          // 0..31 (wave32)
    const float INF = 3.0e38f;

    // ---- Stage x = pred[b] (512x2 f32 = 4 KB) into LDS via async DMA ----
    __shared__ float sX[DTW_N * 2];
    const char* xg = (const char*)(pred + (size_t)b * DTW_N * 2);
    // Flat address of LDS: low 32 bits are the LDS byte address (aperture form).
    uint32_t sx_base = (uint32_t)(uintptr_t)(&sX[0]);
    #pragma unroll
    for (int i = 0; i < 8; ++i) {          // 8 * 32 lanes * 16 B = 4 KB
        uint32_t off = (uint32_t)(i * 32 + lane) * 16u;
        async_b128_to_lds(sx_base + off, xg + off);
    }

    // ---- y columns for this lane stay in registers (32 floats) ----
    const float* yb = target + (size_t)b * DTW_N * 2 + (size_t)lane * COLS_PER_LANE * 2;
    float y0[COLS_PER_LANE], y1[COLS_PER_LANE];
    #pragma unroll
    for (int k = 0; k < COLS_PER_LANE; ++k) {
        y0[k] = yb[2 * k];
        y1[k] = yb[2 * k + 1];
    }

    wait_asynccnt0();                      // x tile now resident in LDS

    // ---- Register-resident DP state ----
    float prev[COLS_PER_LANE];             // D[r-1][j] for own columns
    float cur[COLS_PER_LANE];              // D[r][j]
    #pragma unroll
    for (int k = 0; k < COLS_PER_LANE; ++k) { prev[k] = INF; cur[k] = INF; }

    float bLeft = INF;                     // D[r  ][j0-1] (from lane-1, step s-1)
    float bDiag = INF;                     // D[r-1][j0-1] (from lane-1, step s-2)

    const int NSTEP = DTW_N + 31;          // 543 skewed steps
    for (int s = 0; s < NSTEP; ++s) {
        const int r = s - lane;
        const bool active = (r >= 0) && (r < DTW_N);

        if (active) {
            const float x0 = sX[2 * r];
            const float x1 = sX[2 * r + 1];
            const float left0 = (lane == 0) ? INF : bLeft;
            const float diag0 = (lane == 0) ? INF : bDiag;

            if (r == 0) {
                // Row 0 is a running cumsum of costs.
                float run = left0;
                #pragma unroll
                for (int k = 0; k < COLS_PER_LANE; ++k) {
                    const float c = fabsf(x0 - y0[k]) + fabsf(x1 - y1[k]);
                    run = (lane == 0 && k == 0) ? c : (c + run);
                    cur[k] = run;
                }
            } else {
                float lft = left0;         // D[r][j-1]
                float dgn = diag0;         // D[r-1][j-1]
                #pragma unroll
                for (int k = 0; k < COLS_PER_LANE; ++k) {
                    const float up = prev[k];
                    const float c  = fabsf(x0 - y0[k]) + fabsf(x1 - y1[k]);
                    const float v  = c + fminf(fminf(up, dgn), lft);
                    cur[k] = v;
                    dgn = up;              // next diag = D[r-1][j]
                    lft = v;               // next left = D[r][j]
                }
            }
            #pragma unroll
            for (int k = 0; k < COLS_PER_LANE; ++k) prev[k] = cur[k];
        }

        // Cross-lane boundary exchange (all lanes participate; ds_bpermute).
        const float nb = __shfl_up(cur[COLS_PER_LANE - 1], 1, 32);
        bDiag = bLeft;                     // lane-1's cur[15] from step s-1
        bLeft = nb;                        // lane-1's cur[15] from step s
    }

    // Lane 31's last active step (s = 542) produced D[511][511].
    if (lane == 31) ws[b] = cur[COLS_PER_LANE - 1];
}

__global__ void __launch_bounds__(32)
dtw_mean_kernel(const float* __restrict__ ws, float* __restrict__ out)
{
    const int t = threadIdx.x;             // 32 threads, one wave
    float v = ws[t] + ws[t + 32];
    #pragma unroll
    for (int off = 16; off > 0; off >>= 1)
        v += __shfl_down(v, off, 32);
    if (t == 0) out[0] = v * (1.0f / (float)DTW_B);
}

extern "C" void kernel_launch(void* const* d_in, const int* in_sizes, int n_in,
                              void* d_out, int out_size, void* d_ws, size_t ws_size,
                              hipStream_t stream) {
    (void)in_sizes; (void)n_in; (void)out_size; (void)ws_size;
    const float* pred   = (const float*)d_in[0];   // (64, 512, 2) f32
    const float* target = (const float*)d_in[1];   // (64, 512, 2) f32
    float* ws  = (float*)d_ws;                     // 64 f32 partials
    float* out = (float*)d_out;                    // 1 f32

    dtw_batch_kernel<<<DTW_B, 32, 0, stream>>>(pred, target, ws);
    dtw_mean_kernel<<<1, 32, 0, stream>>>(ws, out);
}